// MultiScaleLDSA_LowMemory_20100446945689
// MI455X (gfx1250) — compile-verified
//
#include <hip/hip_runtime.h>
#include <math.h>

#define NFEAT 512
#define SEQT  1000
#define NB    4
#define NHPS  2
#define MROWS (NB * SEQT)   // 4000

// GEMM tiling
#define BM 32               // block tile M (2 wmma tiles)
#define BN 256              // block tile N (8 waves x 2 wmma tiles)
#define KC 16               // K chunk staged per async copy
#define AS_STRIDE 20        // 16 + pad (multiple of 4 -> 16B-aligned rows)
#define BS_STRIDE 260       // 256 + pad (multiple of 4 -> 16B-aligned rows)

typedef __attribute__((ext_vector_type(2))) float v2f;
typedef __attribute__((ext_vector_type(8))) float v8f;

// 32-bit LDS offset of a __shared__ object (generic -> AS(3) -> int)
#define LDS_OFF(p) ((unsigned)(uintptr_t)(__attribute__((address_space(3))) const void*)(p))

// ---------------------------------------------------------------------------
// Tiled fp32 WMMA GEMM:  C[M,N] = A[M,K] * B[K,N]   (optionally ReLU on C)
// 256 threads = 8 waves. Block tile 32x256; each wave owns a 32x32 register
// tile (4 x v8f accumulators), so every A/B fragment feeds 2 WMMAs.
// K staged in 16-wide chunks via GLOBAL_LOAD_ASYNC_TO_LDS_B128 (ASYNCcnt),
// double-buffered: chunk k+1 DMA overlaps chunk k's 16 v_wmma_f32_16x16x4.
// Requires M%32==0, N%256==0, K%16==0 (M=4000, N=K=512 here).
// ---------------------------------------------------------------------------
template <bool RELU>
__global__ __launch_bounds__(256) void gemm_f32_wmma(const float* __restrict__ A,
                                                     const float* __restrict__ Bw,
                                                     float* __restrict__ C,
                                                     int M, int N, int K) {
    __shared__ float As[2][BM][AS_STRIDE];   // 32 x 16 tile (+pad)
    __shared__ float Bs[2][KC][BS_STRIDE];   // 16 x 256 tile (+pad)

    const int tid  = threadIdx.x;
    const int lane = tid & 31;
    const int wave = tid >> 5;               // 0..7 -> n subtile pair
    const int m0   = blockIdx.x * BM;
    const int n0   = blockIdx.y * BN;

    // WMMA f32 16x16x4 fragment lane mapping (ISA 7.12.2):
    //  A: lane L, vgpr j -> A[L&15][2*(L>>4)+j]
    //  B: lane L, vgpr j -> B[2*(L>>4)+j][L&15]
    //  D: lane L, vgpr r -> D[r + 8*(L>>4)][L&15]
    const int lm    = lane & 15;
    const int khalf = (lane >> 4) * 2;
    const int nA    = wave * 32;             // wave's n origin inside block tile

    const float* Aptr = A;
    const float* Bptr = Bw;

    // async-staging thread assignments (float4 granularity, coalesced)
    const int a_r  = tid >> 2;               // 0..63 (only tid<128 used: rows 0..31)
    const int a_c4 = (tid & 3) << 2;         // 0,4,8,12

    // Issue async DMA of one K-chunk into buffer `buf`
    auto issue = [&](int buf, int k0) {
        if (tid < (BM * KC / 4)) {           // 128 lanes: 32x16 A tile
            unsigned lds  = LDS_OFF(&As[buf][a_r][a_c4]);
            unsigned voff = (unsigned)(((m0 + a_r) * K + k0 + a_c4) * 4);
            asm volatile("global_load_async_to_lds_b128 %0, %1, %2"
                         :: "v"(lds), "v"(voff), "s"(Aptr) : "memory");
        }
#pragma unroll
        for (int p = 0; p < 4; ++p) {        // 16x256 B tile: 1024 float4 / 256 thr
            int gidx = tid + p * 256;
            int r    = gidx >> 6;
            int c4   = (gidx & 63) << 2;
            unsigned lds  = LDS_OFF(&Bs[buf][r][c4]);
            unsigned voff = (unsigned)(((k0 + r) * N + n0 + c4) * 4);
            asm volatile("global_load_async_to_lds_b128 %0, %1, %2"
                         :: "v"(lds), "v"(voff), "s"(Bptr) : "memory");
        }
    };

    v8f acc00 = {}, acc01 = {}, acc10 = {}, acc11 = {};

    const int nch = K / KC;
    issue(0, 0);
    for (int ch = 0; ch < nch; ++ch) {
        asm volatile("s_wait_asynccnt 0x0" ::: "memory");   // my wave's DMAs landed
        __syncthreads();                                     // everyone's DMAs visible
        if (ch + 1 < nch) issue((ch & 1) ^ 1, (ch + 1) * KC);

        const int buf = ch & 1;
#pragma unroll
        for (int kk = 0; kk < KC; kk += 4) {
            v2f a0, a1, b0, b1;
            a0.x = As[buf][lm][kk + khalf];
            a0.y = As[buf][lm][kk + khalf + 1];
            a1.x = As[buf][16 + lm][kk + khalf];
            a1.y = As[buf][16 + lm][kk + khalf + 1];
            b0.x = Bs[buf][kk + khalf][nA + lm];
            b0.y = Bs[buf][kk + khalf + 1][nA + lm];
            b1.x = Bs[buf][kk + khalf][nA + 16 + lm];
            b1.y = Bs[buf][kk + khalf + 1][nA + 16 + lm];
            acc00 = __builtin_amdgcn_wmma_f32_16x16x4_f32(false, a0, false, b0, (short)0, acc00, false, false);
            acc01 = __builtin_amdgcn_wmma_f32_16x16x4_f32(false, a0, false, b1, (short)0, acc01, false, false);
            acc10 = __builtin_amdgcn_wmma_f32_16x16x4_f32(false, a1, false, b0, (short)0, acc10, false, false);
            acc11 = __builtin_amdgcn_wmma_f32_16x16x4_f32(false, a1, false, b1, (short)0, acc11, false, false);
        }
        __syncthreads();                                     // chunk fully consumed
    }

    // store 32x32 wave tile (lanes 0..15 / 16..31 write 64B-contiguous runs)
    const int ncol = n0 + nA + lm;
#pragma unroll
    for (int r = 0; r < 8; ++r) {
        int mrow = m0 + r + (lane >> 4) * 8;
        float v00 = acc00[r], v01 = acc01[r], v10 = acc10[r], v11 = acc11[r];
        if (RELU) {
            v00 = fmaxf(v00, 0.0f); v01 = fmaxf(v01, 0.0f);
            v10 = fmaxf(v10, 0.0f); v11 = fmaxf(v11, 0.0f);
        }
        C[mrow * N + ncol]            = v00;
        C[mrow * N + ncol + 16]       = v01;
        C[(mrow + 16) * N + ncol]     = v10;
        C[(mrow + 16) * N + ncol + 16]= v11;
    }
}

// ---------------------------------------------------------------------------
// Per-row: w = relu_q_row(512) @ W2(512, 2c), then per-head softmax over c.
// One block (256 threads) per row. q row staged in LDS.
// ---------------------------------------------------------------------------
__global__ __launch_bounds__(256) void attn_softmax_kernel(const float* __restrict__ Q,
                                                           const float* __restrict__ W2,
                                                           float* __restrict__ attn,
                                                           int c) {
    __shared__ float qrow[NFEAT];
    __shared__ float wv[192];       // up to 2c = 162
    __shared__ float red[4];        // max[h], sum[h]

    const int tid  = threadIdx.x;
    const int row  = blockIdx.x;
    const int twoc = 2 * c;

    for (int i = tid; i < NFEAT; i += 256) qrow[i] = Q[row * NFEAT + i];
    __syncthreads();

    if (tid < twoc) {
        float s = 0.0f;
        for (int k = 0; k < NFEAT; ++k) s += qrow[k] * W2[k * twoc + tid];
        wv[tid] = s;
    }
    __syncthreads();

    if (tid < NHPS) {
        float m = -1e30f;
        for (int j = 0; j < c; ++j) m = fmaxf(m, wv[tid * c + j]);
        float s = 0.0f;
        for (int j = 0; j < c; ++j) s += __expf(wv[tid * c + j] - m);
        red[tid]     = m;
        red[2 + tid] = s;
    }
    __syncthreads();

    if (tid < twoc) {
        int h = tid / c;
        attn[row * twoc + tid] = __expf(wv[tid] - red[h]) / red[2 + h];
    }
}

// ---------------------------------------------------------------------------
// Sliding-window weighted sum: acc[row,f] += sw[i] * sum_j attn[row,h,j] * V[row-N+j, f]
// One block (256 threads) per row; thread owns f=tid (head 0) and f=tid+256 (head 1).
// ---------------------------------------------------------------------------
__global__ __launch_bounds__(256) void window_accum_kernel(const float* __restrict__ attn,
                                                           const float* __restrict__ V,
                                                           const float* __restrict__ scale_w,
                                                           float* __restrict__ acc,
                                                           int c, int scale_idx) {
    __shared__ float a_sh[192];

    const int tid  = threadIdx.x;
    const int row  = blockIdx.x;
    const int twoc = 2 * c;
    const int b    = row / SEQT;
    const int t    = row - b * SEQT;
    const int Npad = (c - 1) / 2;

    if (tid < twoc) a_sh[tid] = attn[row * twoc + tid];
    __syncthreads();

    // softmax over the 4 scale weights, pick scale_idx
    float s0 = scale_w[0], s1 = scale_w[1], s2 = scale_w[2], s3 = scale_w[3];
    float mx = fmaxf(fmaxf(s0, s1), fmaxf(s2, s3));
    float e0 = __expf(s0 - mx), e1 = __expf(s1 - mx);
    float e2 = __expf(s2 - mx), e3 = __expf(s3 - mx);
    float es = e0 + e1 + e2 + e3;
    float swi = ((scale_idx == 0) ? e0 : (scale_idx == 1) ? e1
                : (scale_idx == 2) ? e2 : e3) / es;

    const int f0 = tid;          // head 0, d = tid
    const int f1 = tid + 256;    // head 1, d = tid
    float x0 = 0.0f, x1 = 0.0f;

    for (int j = 0; j < c; ++j) {
        int st = t - Npad + j;
        if (st < 0 || st >= SEQT) continue;
        const float* vrow = V + (b * SEQT + st) * NFEAT;
        x0 += a_sh[j]     * vrow[f0];
        x1 += a_sh[c + j] * vrow[f1];
    }
    acc[row * NFEAT + f0] += swi * x0;
    acc[row * NFEAT + f1] += swi * x1;
}

// ---------------------------------------------------------------------------
extern "C" void kernel_launch(void* const* d_in, const int* in_sizes, int n_in,
                              void* d_out, int out_size, void* d_ws, size_t ws_size,
                              hipStream_t stream) {
    const float* query    = (const float*)d_in[0];
    // d_in[1] = key (unused by reference math)
    const float* value    = (const float*)d_in[2];
    const float* w1       = (const float*)d_in[3];   // (4,512,512)
    const float* w3       = (const float*)d_in[4];   // (4,512,512)
    const float* scale_w  = (const float*)d_in[5];   // (4,)
    const float* w_out    = (const float*)d_in[6];   // (512,512)
    const float* w2s[4]   = { (const float*)d_in[7], (const float*)d_in[8],
                              (const float*)d_in[9], (const float*)d_in[10] };
    float* out = (float*)d_out;

    const int contexts[4] = {21, 41, 61, 81};

    // workspace layout (fp32)
    float* accb  = (float*)d_ws;                 // 4000*512
    float* Qb    = accb  + MROWS * NFEAT;        // 4000*512
    float* Vb    = Qb    + MROWS * NFEAT;        // 4000*512
    float* attnb = Vb    + MROWS * NFEAT;        // 4000*162 max

    hipMemsetAsync(accb, 0, (size_t)MROWS * NFEAT * sizeof(float), stream);

    dim3 gemmGrid(MROWS / BM, NFEAT / BN);   // 125 x 2
    dim3 gemmBlk(256);

    for (int i = 0; i < 4; ++i) {
        const int c = contexts[i];
        // Q = relu(query @ w1[i]), V = value @ w3[i]
        gemm_f32_wmma<true><<<gemmGrid, gemmBlk, 0, stream>>>(
            query, w1 + (size_t)i * NFEAT * NFEAT, Qb, MROWS, NFEAT, NFEAT);
        gemm_f32_wmma<false><<<gemmGrid, gemmBlk, 0, stream>>>(
            value, w3 + (size_t)i * NFEAT * NFEAT, Vb, MROWS, NFEAT, NFEAT);
        // attn = softmax_per_head(Q @ w2_i)
        attn_softmax_kernel<<<MROWS, 256, 0, stream>>>(Qb, w2s[i], attnb, c);
        // acc += sw[i] * window(attn, V)
        window_accum_kernel<<<MROWS, 256, 0, stream>>>(attnb, Vb, scale_w, accb, c, i);
    }

    // out = acc @ w_out
    gemm_f32_wmma<false><<<gemmGrid, gemmBlk, 0, stream>>>(
        accb, w_out, out, MROWS, NFEAT, NFEAT);
}